// PoseSkeleton_28905129902886
// MI455X (gfx1250) — compile-verified
//
#include <hip/hip_runtime.h>
#include <hip/hip_bf16.h>
#include <stdint.h>

#define NB  65536
#define NK  24
#define TPB 256

// One lane = one batch element. LDS slot (j, lane) = 13 dwords (12 data + pad;
// stride 13 -> gcd(13,64)=1 -> bank-conflict-free both ways).
// Phase A (async engine writes): slot j holds input R_j (9) + pos_j (3).
// Phase B (after consume):      slot j holds record R_acc_j (9) + v_j (3),
//   where v = t_acc - R_acc*pos is closed under the recursion:
//     t_acc_j = R_acc_p * pos_j + v_p
//     R_acc_j = R_acc_p * R_j
//     v_j     = t_acc_j - R_acc_j * pos_j
//   so children never need the parent's raw position.
__global__ __launch_bounds__(TPB) void pose_skel_kernel(
    const float* __restrict__ rot,   // (B,K,3,3)
    const float* __restrict__ pos,   // (B,K,3)
    const int*   __restrict__ par,   // (K)
    float* __restrict__ outT,        // (B,K,4,4)
    float* __restrict__ outP)        // (B,K,3)
{
  __shared__ float sT[NK * TPB * 13];   // 319,488 B of the 320 KB WGP LDS

  const int tid = threadIdx.x;
  const int b   = blockIdx.x * TPB + tid;

  const uint32_t base = (uint32_t)(uintptr_t)(&sT[0]);

  // Async-stage joint j's R (9 floats) + pos (3 floats) into slot j.
  // Global and LDS element strides are both 4B, so one base VGPR pair +
  // offset:N drives both sides (INST_OFFSET applies to dsaddr AND memaddr).
  auto stage = [&](int j) {
    const float* ra = rot + ((size_t)b * NK + j) * 9;
    const float* pa = pos + ((size_t)b * NK + j) * 3;
    uint32_t l0 = base + (uint32_t)(j * TPB + tid) * 52u;  // 13 dwords/slot
    uint32_t l1 = l0 + 36u;
    asm volatile(
        "global_load_async_to_lds_b32 %0, %1, off\n\t"
        "global_load_async_to_lds_b32 %0, %1, off offset:4\n\t"
        "global_load_async_to_lds_b32 %0, %1, off offset:8\n\t"
        "global_load_async_to_lds_b32 %0, %1, off offset:12\n\t"
        "global_load_async_to_lds_b32 %0, %1, off offset:16\n\t"
        "global_load_async_to_lds_b32 %0, %1, off offset:20\n\t"
        "global_load_async_to_lds_b32 %0, %1, off offset:24\n\t"
        "global_load_async_to_lds_b32 %0, %1, off offset:28\n\t"
        "global_load_async_to_lds_b32 %0, %1, off offset:32\n\t"
        :
        : "v"(l0), "v"(ra)
        : "memory");
    asm volatile(
        "global_load_async_to_lds_b32 %0, %1, off\n\t"
        "global_load_async_to_lds_b32 %0, %1, off offset:4\n\t"
        "global_load_async_to_lds_b32 %0, %1, off offset:8\n\t"
        :
        : "v"(l1), "v"(pa)
        : "memory");
  };

  // Prime a 5-slot prefetch window: 60 async loads in flight (cap is 63).
#pragma unroll
  for (int j = 0; j < 5; ++j) stage(j);

#pragma unroll
  for (int j = 0; j < NK; ++j) {
    // Async loads complete in order: <= 12*min(4, 23-j) outstanding means
    // slot j's 12 loads have landed while the window stays in flight.
    if (j < 20)      asm volatile("s_wait_asynccnt 48" ::: "memory");
    else if (j == 20) asm volatile("s_wait_asynccnt 36" ::: "memory");
    else if (j == 21) asm volatile("s_wait_asynccnt 24" ::: "memory");
    else if (j == 22) asm volatile("s_wait_asynccnt 12" ::: "memory");
    else              asm volatile("s_wait_asynccnt 0"  ::: "memory");

    if (j + 5 < NK) stage(j + 5);   // refill window (48 + 12 <= 63, no stall)

    float* slot = &sT[(size_t)(j * TPB + tid) * 13];
    float R[9], P[3];
#pragma unroll
    for (int c = 0; c < 9; ++c) R[c] = slot[c];
#pragma unroll
    for (int c = 0; c < 3; ++c) P[c] = slot[9 + c];

    float Ra[9], v[3], ta[3];
    if (j == 0) {
#pragma unroll
      for (int c = 0; c < 9; ++c) Ra[c] = R[c];
#pragma unroll
      for (int i = 0; i < 3; ++i) {
        ta[i] = P[i];
        v[i]  = P[i] - (Ra[3*i]*P[0] + Ra[3*i+1]*P[1] + Ra[3*i+2]*P[2]);
      }
    } else {
      const int p = par[j];                          // uniform -> s_load
      const float* pr = &sT[(size_t)(p * TPB + tid) * 13];
      float Rp[9], vp[3];
#pragma unroll
      for (int c = 0; c < 9; ++c) Rp[c] = pr[c];
#pragma unroll
      for (int c = 0; c < 3; ++c) vp[c] = pr[9 + c];
#pragma unroll
      for (int i = 0; i < 3; ++i)
        ta[i] = Rp[3*i]*P[0] + Rp[3*i+1]*P[1] + Rp[3*i+2]*P[2] + vp[i];
#pragma unroll
      for (int i = 0; i < 3; ++i)
#pragma unroll
        for (int k = 0; k < 3; ++k)
          Ra[3*i+k] = Rp[3*i]*R[k] + Rp[3*i+1]*R[3+k] + Rp[3*i+2]*R[6+k];
#pragma unroll
      for (int i = 0; i < 3; ++i)
        v[i] = ta[i] - (Ra[3*i]*P[0] + Ra[3*i+1]*P[1] + Ra[3*i+2]*P[2]);
    }

    // Overwrite slot j with the record for children (same-lane DS ops are
    // in-order, so input reads above are safely ordered before this).
#pragma unroll
    for (int c = 0; c < 9; ++c) slot[c] = Ra[c];
#pragma unroll
    for (int c = 0; c < 3; ++c) slot[9 + c] = v[c];

    // joint_transforms[b,j] = [Ra | v ; 0 0 0 1]  (64B-aligned float4 stores)
    float4* o = reinterpret_cast<float4*>(outT + ((size_t)b * NK + j) * 16);
    o[0] = make_float4(Ra[0], Ra[1], Ra[2], v[0]);
    o[1] = make_float4(Ra[3], Ra[4], Ra[5], v[1]);
    o[2] = make_float4(Ra[6], Ra[7], Ra[8], v[2]);
    o[3] = make_float4(0.f, 0.f, 0.f, 1.f);
    // joint_positions_posed[b,j] = t_acc
    float* op = outP + ((size_t)b * NK + j) * 3;
    op[0] = ta[0]; op[1] = ta[1]; op[2] = ta[2];
  }
}

extern "C" void kernel_launch(void* const* d_in, const int* in_sizes, int n_in,
                              void* d_out, int out_size, void* d_ws, size_t ws_size,
                              hipStream_t stream) {
  (void)in_sizes; (void)n_in; (void)out_size; (void)d_ws; (void)ws_size;
  const float* rot = (const float*)d_in[0];
  const float* pos = (const float*)d_in[1];
  const int*   par = (const int*)d_in[2];
  float* outT = (float*)d_out;
  float* outP = outT + (size_t)NB * NK * 16;
  pose_skel_kernel<<<dim3(NB / TPB), dim3(TPB), 0, stream>>>(rot, pos, par, outT, outP);
}